// ConvHex_58231166599274
// MI455X (gfx1250) — compile-verified
//
#include <hip/hip_runtime.h>
#include <math.h>

typedef __attribute__((ext_vector_type(2))) float v2f;
typedef __attribute__((ext_vector_type(8))) float v8f;

#define BSZ   128
#define CIN   32
#define COUT  32
#define HN    4096
#define K7    7
#define KTOT  224                    // CIN * K7
#define NTOT  (BSZ*COUT*HN)          // 16,777,216
#define BH    (BSZ*HN)               // 524,288 per-channel BN count
#define BLK   96                     // dwords per K-pair block in wLds (bank-conflict-free)

// ws layout (bytes):
//   [  0,  16) : double g[2]  {sum, sumsq} of x (global std)
//   [ 16, 272) : float cs[64] {mean[32], invstd[32]} after finalize
//   [512,16896): float slots[64][64] BN partial accumulators
#define WS_SLOTS_OFF_F (512/4)
#define ZERO_FLOATS 4224             // zero bytes [0, 16896)

__global__ __launch_bounds__(256) void k_init(float* ws) {
  for (int i = threadIdx.x; i < ZERO_FLOATS; i += 256) ws[i] = 0.0f;
}

// ---------- global std: sum & sumsq over all of x ----------
__global__ __launch_bounds__(256)
void k_gstats(const float4* __restrict__ x4, double* __restrict__ g) {
  double s = 0.0, q = 0.0;
  const int n4 = NTOT / 4;
  for (int i = blockIdx.x*blockDim.x + threadIdx.x; i < n4; i += gridDim.x*blockDim.x) {
    float4 v = x4[i];
    s += (double)((v.x + v.y) + (v.z + v.w));
    q += (double)(v.x*v.x + v.y*v.y) + (double)(v.z*v.z + v.w*v.w);
  }
  __shared__ double ls[256], lq[256];
  ls[threadIdx.x] = s; lq[threadIdx.x] = q;
  __syncthreads();
  for (int off = 128; off > 0; off >>= 1) {
    if (threadIdx.x < off) {
      ls[threadIdx.x] += ls[threadIdx.x + off];
      lq[threadIdx.x] += lq[threadIdx.x + off];
    }
    __syncthreads();
  }
  if (threadIdx.x == 0) { atomicAdd(&g[0], ls[0]); atomicAdd(&g[1], lq[0]); }
}

// ---------- main hex-conv via fp32 WMMA ----------
// K-axis is tap-major (p = k*32 + i): per 4-wide K-step the neighbor column
// and mask are invariant; global/DS offsets are immediates.  Weights stored
// as K-pair blocks so each WMMA B fragment is ONE aligned ds_load_b64 into
// an adjacent VGPR pair (no reassembly movs), conflict-free banking.
// grid: (HN/16) * (BSZ/4) = 8192 blocks of 128 threads (4 waves).
__global__ __launch_bounds__(128)
void k_conv(const float* __restrict__ x, const float* __restrict__ wgt,
            const float* __restrict__ bias, const int* __restrict__ nbr,
            const double* __restrict__ g, float* __restrict__ out,
            float* __restrict__ slots) {
  __shared__ float wLds[(KTOT/2)*BLK];       // 112 K-pair blocks, 42 KB
  __shared__ int   ncol[16*8];               // per-hex tap columns (-1 invalid)
  __shared__ float fmul[16];                 // (1/std) / (valid+1e-6) per hex row
  __shared__ float blkSum[COUT], blkSq[COUT];
  __shared__ float outStage[4][2][16][20];   // padded transpose staging (10 KB)

  const int tid = threadIdx.x;
  const int h0  = (blockIdx.x & 255) << 4;
  const int b0  = (blockIdx.x >> 8) << 2;

  // stage weights: coalesced global read, scatter to K-pair layout:
  // element (p = k*32+i, o)  ->  wLds[(p>>1)*BLK + o*2 + (p&1)]
  for (int idx = tid; idx < KTOT*COUT; idx += 128) {
    int o   = idx / KTOT;
    int ikf = idx - o*KTOT;                  // = i*7 + k
    int i   = ikf / K7;
    int k   = ikf - i*K7;
    int p   = k*CIN + i;
    wLds[(p >> 1)*BLK + o*2 + (p & 1)] = wgt[idx];
  }
  if (tid < 16) {
    int m = tid, tv = 1;
    ncol[m*8 + 0] = h0 + m;                  // self tap (k=0)
    #pragma unroll
    for (int k = 1; k < K7; ++k) {
      int c = nbr[(h0 + m)*(K7 - 1) + (k - 1)];
      ncol[m*8 + k] = c;
      tv += (c >= 0) ? 1 : 0;
    }
    double sum = g[0], sq = g[1];
    double N = (double)NTOT;
    double var = (sq - sum*sum/N) / (N - 1.0);            // ddof=1
    float  s   = (float)(1.0 / (sqrt(var) + 1e-5));
    fmul[m] = s / ((float)tv + 1e-6f);
  }
  if (tid < 32) { blkSum[tid] = 0.0f; blkSq[tid] = 0.0f; }
  __syncthreads();

  const int wv    = tid >> 5;
  const int lane  = tid & 31;
  const int m     = lane & 15;     // A row / B column within 16-tile
  const int hi    = lane >> 4;     // lane half selects K pair
  const int hioff = hi * 2;
  const int b     = b0 + wv;
  const float* xb = x + (size_t)b * CIN * HN + hioff * HN;
  // per-lane LDS fragment base: K-pair pp = k*16 + s*2 + hi, lane column m
  const float* __restrict__ wbase = &wLds[hi*BLK + m*2];

  v8f c0 = {}; v8f c1 = {};

  #pragma unroll
  for (int k = 0; k < K7; ++k) {
    const int   cc   = ncol[m*8 + k];
    const float mskf = (cc >= 0) ? 1.0f : 0.0f;
    const int   cp   = (cc > 0) ? cc : 0;
    const float* __restrict__ xp = xb + cp;                // lane base for tap k
    #pragma unroll
    for (int s = 0; s < 8; ++s) {
      v2f a;
      a.x = xp[(s*4)    * HN] * mskf;                      // imm offset s*65536B
      a.y = xp[(s*4 + 1)* HN] * mskf;                      // imm offset +16384B
      const float* wp = wbase + (k*16 + s*2)*BLK;          // imm DS offsets
      v2f bb0 = *(const v2f*)(wp);                         // ds_load_b64
      v2f bb1 = *(const v2f*)(wp + 32);                    // ds_load_b64 (+128B)
      c0 = __builtin_amdgcn_wmma_f32_16x16x4_f32(false, a, false, bb0, (short)0, c0, false, false);
      c1 = __builtin_amdgcn_wmma_f32_16x16x4_f32(false, a, false, bb1, (short)0, c1, false, false);
    }
  }

  // epilogue: scale per-hex, add bias, BN partial stats, transpose via LDS
  const float bo0 = bias[m], bo1 = bias[m + 16];
  float ps0 = 0.f, pq0 = 0.f, ps1 = 0.f, pq1 = 0.f;
  #pragma unroll
  for (int v = 0; v < 8; ++v) {
    int   mr = v + hi*8;                 // D row (hex within tile)
    float fv = fmul[mr];
    float y0 = c0[v]*fv + bo0;
    float y1 = c1[v]*fv + bo1;
    ps0 += y0; pq0 += y0*y0;
    ps1 += y1; pq1 += y1*y1;
    outStage[wv][0][m][mr] = y0;
    outStage[wv][1][m][mr] = y1;
  }
  atomicAdd(&blkSum[m],      ps0); atomicAdd(&blkSq[m],      pq0);
  atomicAdd(&blkSum[m + 16], ps1); atomicAdd(&blkSq[m + 16], pq1);
  __syncthreads();

  { // lane r -> channel r, contiguous 16 floats along H => b128 stores
    int t = lane >> 4, o = lane & 15;
    const float* src = &outStage[wv][t][o][0];
    float* dst = out + (((size_t)b*COUT + lane)*HN + h0);
    #pragma unroll
    for (int jj = 0; jj < 4; ++jj)
      *(float4*)(dst + jj*4) = *(const float4*)(src + jj*4);
  }
  if (tid < 64) {
    float val = (tid < 32) ? blkSum[tid] : blkSq[tid - 32];
    atomicAdd(&slots[(blockIdx.x & 63)*64 + tid], val);
  }
}

// ---------- finalize per-channel mean / invstd ----------
__global__ __launch_bounds__(32)
void k_final(const float* __restrict__ slots, float* __restrict__ cs) {
  int t = threadIdx.x;                     // 0..31 channel
  float s = 0.f, q = 0.f;
  for (int sl = 0; sl < 64; ++sl) {
    s += slots[sl*64 + t];
    q += slots[sl*64 + 32 + t];
  }
  float mean = s / (float)BH;
  float var  = q / (float)BH - mean*mean;  // biased var
  cs[t]      = mean;
  cs[32 + t] = rsqrtf(var + 1e-5f);
}

// ---------- apply BatchNorm (vectorized in-place pass) ----------
__global__ __launch_bounds__(256)
void k_bn(float4* __restrict__ out4, const float* __restrict__ cs,
          const float* __restrict__ gamma, const float* __restrict__ beta) {
  const int n4 = NTOT / 4;
  for (int i = blockIdx.x*blockDim.x + threadIdx.x; i < n4; i += gridDim.x*blockDim.x) {
    int o = (i >> 10) & 31;                // HN/4 = 1024 float4 per (b,o) row
    float ga = gamma[o] * cs[32 + o];
    float be = beta[o] - cs[o] * ga;
    float4 v = out4[i];
    v.x = v.x*ga + be; v.y = v.y*ga + be;
    v.z = v.z*ga + be; v.w = v.w*ga + be;
    out4[i] = v;
  }
}

extern "C" void kernel_launch(void* const* d_in, const int* in_sizes, int n_in,
                              void* d_out, int out_size, void* d_ws, size_t ws_size,
                              hipStream_t stream) {
  const float* x     = (const float*)d_in[0];
  const float* wgt   = (const float*)d_in[1];
  const float* bias  = (const float*)d_in[2];
  const float* gamma = (const float*)d_in[3];
  const float* beta  = (const float*)d_in[4];
  const int*   nbr   = (const int*)d_in[5];   // neighbors [H][6]
  // d_in[6] = groups (==1), unused
  float*  out   = (float*)d_out;
  float*  ws    = (float*)d_ws;
  double* g     = (double*)d_ws;
  float*  cs    = ws + 4;                 // byte offset 16
  float*  slots = ws + WS_SLOTS_OFF_F;    // byte offset 512

  k_init  <<<1,    256, 0, stream>>>(ws);
  k_gstats<<<2048, 256, 0, stream>>>((const float4*)x, g);
  k_conv  <<<8192, 128, 0, stream>>>(x, wgt, bias, nbr, g, out, slots);
  k_final <<<1,     32, 0, stream>>>(slots, cs);
  k_bn    <<<8192, 256, 0, stream>>>((float4*)out, cs, gamma, beta);
}